// MambaBlock_23029614641550
// MI455X (gfx1250) — compile-verified
//
#include <hip/hip_runtime.h>
#include <math.h>

// ---------------- problem constants ----------------
#define DMODEL 1024
#define DSTATE 16
#define DCONV  4
#define DINNER 2048
#define DTRANK 64
#define BATCH  2
#define SEQLEN 4096
#define MROWS  (BATCH*SEQLEN)      // 8192
#define NCHUNK 64
#define CHUNK  (SEQLEN/NCHUNK)     // 64
#define XDBL_W (DTRANK + 2*DSTATE) // 96

typedef __attribute__((ext_vector_type(2))) float v2f;
typedef __attribute__((ext_vector_type(8))) float v8f;
typedef __attribute__((ext_vector_type(4))) int   v4i;

// ---- gfx1250 async global->LDS path (guarded: falls back to sync staging) ----
#if __has_builtin(__builtin_amdgcn_global_load_async_to_lds_b128)
#define HAVE_ASYNC_LDS 1
#if __has_builtin(__builtin_amdgcn_s_wait_asynccnt)
#define WAIT_ASYNC(n) __builtin_amdgcn_s_wait_asynccnt(n)
#else
#define WAIT_ASYNC(n) asm volatile("s_wait_asynccnt %0" :: "n"(n) : "memory")
#endif
// flat->LDS: hardware maps flat LDS addresses by truncation
__device__ __forceinline__ __attribute__((address_space(3))) v4i* as_lds_v4(void* p) {
    return (__attribute__((address_space(3))) v4i*)(uint32_t)(uintptr_t)p;
}
// flat->global: identical bits in the global aperture
__device__ __forceinline__ __attribute__((address_space(1))) v4i* as_glb_v4(const void* p) {
    return (__attribute__((address_space(1))) v4i*)(uintptr_t)p;
}
#else
#define HAVE_ASYNC_LDS 0
#endif

// =====================================================================
// Blocked f32 WMMA GEMM:  C[m,n] = sum_k A[m,k] * Bw[n,k]  (Bw row-major N x K)
// Block = 256 threads / 8 waves, block tile = 128 x (16*NT).
//  - B panel staged in LDS, shared by all 8 waves.  LDS layout [n][k],
//    row stride KBLK+4 -> conflict-free ds_load_b64 fragments.
//  - DOUBLE-BUFFERED via GLOBAL_LOAD_ASYNC_TO_LDS_B128 (ASYNCcnt): the
//    next K-block's panel streams into buffer p^1 while computing from p.
//    Each wave issues exactly NT async ops per stage, so
//    s_wait_asynccnt(NT) after issuing the next stage => current landed.
//  - A strip for the whole 64-deep K-block burst-loaded to registers.
//  - B fragments pipelined one k-step ahead out of LDS.
// EPI: 0 = plain store, 1 = +bias[n] then softplus.
// =====================================================================
template<int NT, int EPI>
__global__ __launch_bounds__(256)
void wmma_gemm_f32(const float* __restrict__ A, int lda,
                   const float* __restrict__ Bw, int ldb,
                   float* __restrict__ C, int ldc,
                   int M, int N, int K,
                   const float* __restrict__ bias)
{
    constexpr int KBLK = 64;
    constexpr int SR   = KBLK + 4;          // padded LDS row stride (floats)
    constexpr int BUFN = 16 * NT * SR;      // floats per buffer
#if HAVE_ASYNC_LDS
    __shared__ float Bs[2 * BUFN];
#else
    __shared__ float Bs[BUFN];
#endif

    const int tid  = threadIdx.x;
    const int lane = tid & 31;
    const int wave = tid >> 5;
    const int mblocks = M >> 7;             // 128-row blocks
    const int mb = blockIdx.x % mblocks;    // consecutive blocks share B panel
    const int nb = blockIdx.x / mblocks;
    const int m0 = (mb << 7) + (wave << 4);
    const int n0 = nb * (16 * NT);

    // f32 16x16x4 fragment layout (ISA 7.12.2):
    //   A: lane&15 = m, (lane>>4)*2 = k-half; VGPR j holds A[m][khalf+j]
    //   B: lane&15 = n, same k-half split
    //   C/D: VGPR r holds C[r + 8*(lane>=16)][lane&15]
    const int mA    = m0 + (lane & 15);
    const int khalf = (lane >> 4) << 1;     // 0 or 2
    const int nlo   = lane & 15;

    v8f acc[NT];
#pragma unroll
    for (int j = 0; j < NT; ++j) { v8f zz = {0.f,0.f,0.f,0.f,0.f,0.f,0.f,0.f}; acc[j] = zz; }

    const float* Arow = A + (size_t)mA * lda + khalf;

#if HAVE_ASYNC_LDS
    // ---- async stage of one B panel: 256*NT x b128 chunks, NT per thread ----
    auto stage = [&](int kb, int buf) {
#pragma unroll
        for (int it = 0; it < NT; ++it) {
            const int i  = tid + 256 * it;
            const int r  = i >> 4;           // n_local (16 chunks per row)
            const int qf = (i & 15) << 2;    // k offset in floats
            __builtin_amdgcn_global_load_async_to_lds_b128(
                as_glb_v4(Bw + (size_t)(n0 + r) * ldb + kb + qf),
                as_lds_v4(&Bs[buf * BUFN + r * SR + qf]), 0, 0);
        }
    };
    int p = 0;
    stage(0, 0);
#endif

    for (int kb = 0; kb < K; kb += KBLK) {
        // ---- burst-load this wave's A strip for the whole K-block ----
        v2f areg[16];
#pragma unroll
        for (int s = 0; s < 16; ++s)
            areg[s] = *(const v2f*)(Arow + kb + 4 * s);

#if HAVE_ASYNC_LDS
        // ---- kick off next panel, then wait for current one ----
        if (kb + KBLK < K) { stage(kb + KBLK, p ^ 1); WAIT_ASYNC(NT); }
        else               { WAIT_ASYNC(0); }
        __syncthreads();                       // everyone's panel p landed
        const float* Bbase = &Bs[p * BUFN];
#else
        // ---- synchronous staging fallback ----
        __syncthreads();                       // previous iteration's readers done
#pragma unroll 2
        for (int i = tid; i < 16 * NT * (KBLK / 4); i += 256) {
            const int r = i / (KBLK / 4);
            const int q = (i - r * (KBLK / 4)) * 4;
            const float4 v = *(const float4*)(Bw + (size_t)(n0 + r) * ldb + kb + q);
            *(float4*)(&Bs[r * SR + q]) = v;
        }
        __syncthreads();
        const float* Bbase = &Bs[0];
#endif

        // ---- 16 k-steps; B fragments pipelined one step ahead ----
        v2f bcur[NT];
#pragma unroll
        for (int j = 0; j < NT; ++j)
            bcur[j] = *(const v2f*)(&Bbase[(j * 16 + nlo) * SR + khalf]);
#pragma unroll
        for (int s = 0; s < 16; ++s) {
            v2f bnxt[NT];
            if (s < 15) {
#pragma unroll
                for (int j = 0; j < NT; ++j)
                    bnxt[j] = *(const v2f*)(&Bbase[(j * 16 + nlo) * SR + 4 * (s + 1) + khalf]);
            }
#pragma unroll
            for (int j = 0; j < NT; ++j)
                acc[j] = __builtin_amdgcn_wmma_f32_16x16x4_f32(
                    false, areg[s], false, bcur[j], (short)0, acc[j], false, false);
            if (s < 15) {
#pragma unroll
                for (int j = 0; j < NT; ++j) bcur[j] = bnxt[j];
            }
        }
        __syncthreads();                       // reads of this buffer done
#if HAVE_ASYNC_LDS
        p ^= 1;
#endif
    }

    const int mBase = m0 + ((lane >> 4) << 3);       // rows 0..7 or 8..15
#pragma unroll
    for (int j = 0; j < NT; ++j) {
        const int n = n0 + j * 16 + nlo;
        float bv = (EPI == 1) ? bias[n] : 0.0f;
#pragma unroll
        for (int r = 0; r < 8; ++r) {
            float v = acc[j][r];
            if (EPI == 1) {                          // softplus(v + bias)
                v += bv;
                v = (v > 20.0f) ? v : logf(1.0f + __expf(v));
            }
            C[(size_t)(mBase + r) * ldc + n] = v;
        }
    }
}

// =====================================================================
// Depthwise causal conv1d (width 4) + bias + SiLU.  x,xc: [MROWS, DINNER]
// =====================================================================
__global__ __launch_bounds__(256)
void conv_silu_kernel(const float* __restrict__ x, const float* __restrict__ w,
                      const float* __restrict__ bias, float* __restrict__ xc)
{
    const int idx = blockIdx.x * 256 + threadIdx.x;   // MROWS*DINNER threads
    const int c = idx & (DINNER - 1);
    const int m = idx >> 11;
    const int l = m & (SEQLEN - 1);
    float acc = bias[c];
#pragma unroll
    for (int k = 0; k < DCONV; ++k) {
        const int ll = l - (DCONV - 1) + k;
        if (ll >= 0)
            acc += x[(size_t)(m - (DCONV - 1) + k) * DINNER + c] * w[c * DCONV + k];
    }
    xc[(size_t)m * DINNER + c] = acc / (1.0f + __expf(-acc));   // silu
}

// =====================================================================
// Scan phase 1: per-chunk partial scan from h=0 -> chunk-final state S and sum(dt).
// Block = 256 channels of one (batch, chunk). B rows staged in LDS.
// dt/xc loads software-pipelined one timestep ahead.
// =====================================================================
__global__ __launch_bounds__(256)
void scan_phase1(const float* __restrict__ xc, const float* __restrict__ dt,
                 const float* __restrict__ xdbl, const float* __restrict__ A_log,
                 float* __restrict__ S, float* __restrict__ dtsum)
{
    __shared__ float Bs[CHUNK * DSTATE];
    const int bid   = blockIdx.x;              // (b*NCHUNK + chunk)*8 + dgrp
    const int dgrp  = bid & 7;
    const int chunk = (bid >> 3) & (NCHUNK - 1);
    const int b     = bid >> 9;
    const int tid   = threadIdx.x;
    const int d     = dgrp * 256 + tid;
    const int mbase = b * SEQLEN + chunk * CHUNK;

    for (int i = tid; i < CHUNK * DSTATE; i += 256) {
        const int t = i >> 4, n = i & 15;
        Bs[i] = xdbl[(size_t)(mbase + t) * XDBL_W + DTRANK + n];
    }
    __syncthreads();

    float a[DSTATE], h[DSTATE];
#pragma unroll
    for (int n = 0; n < DSTATE; ++n) { a[n] = -__expf(A_log[d * DSTATE + n]); h[n] = 0.f; }
    float ds = 0.f;
    float dtv = dt[(size_t)mbase * DINNER + d];
    float xv  = xc[(size_t)mbase * DINNER + d];
    for (int t = 0; t < CHUNK; ++t) {
        float dtn = 0.f, xn = 0.f;
        if (t + 1 < CHUNK) {                       // prefetch next timestep
            const size_t mn = (size_t)(mbase + t + 1);
            dtn = dt[mn * DINNER + d];
            xn  = xc[mn * DINNER + d];
        }
        const float dtx = dtv * xv;
        ds += dtv;
#pragma unroll
        for (int n = 0; n < DSTATE; ++n)
            h[n] = h[n] * __expf(dtv * a[n]) + dtx * Bs[t * DSTATE + n];
        dtv = dtn; xv = xn;
    }
    const size_t sb = (size_t)(b * NCHUNK + chunk) * DINNER + d;
#pragma unroll
    for (int n = 0; n < DSTATE; ++n) S[sb * DSTATE + n] = h[n];
    dtsum[sb] = ds;
}

// =====================================================================
// Scan phase 2: serial combine across the 64 chunk boundaries.
// One thread per (b,d,n); chunk decay = exp(A * sum(dt over chunk)).
// =====================================================================
__global__ __launch_bounds__(256)
void scan_phase2(const float* __restrict__ S, const float* __restrict__ dtsum,
                 const float* __restrict__ A_log, float* __restrict__ H0)
{
    const int idx = blockIdx.x * 256 + threadIdx.x;   // BATCH*DINNER*DSTATE
    const int n = idx & 15;
    const int d = (idx >> 4) & (DINNER - 1);
    const int b = idx >> 15;
    const float a = -__expf(A_log[d * DSTATE + n]);
    float h = 0.f;
    for (int c = 0; c < NCHUNK; ++c) {
        const size_t sb = (size_t)(b * NCHUNK + c) * DINNER + d;
        H0[sb * DSTATE + n] = h;
        h = h * __expf(a * dtsum[sb]) + S[sb * DSTATE + n];
    }
}

// =====================================================================
// Scan phase 3: re-run each chunk from its true initial state, emit
// y = (C.h + D*xc) * silu(z).  B and C rows staged in LDS.
// =====================================================================
__global__ __launch_bounds__(256)
void scan_phase3(const float* __restrict__ xc, const float* __restrict__ dt,
                 const float* __restrict__ xdbl, const float* __restrict__ A_log,
                 const float* __restrict__ Dp, const float* __restrict__ z,
                 const float* __restrict__ H0, float* __restrict__ y)
{
    __shared__ float Bs[CHUNK * DSTATE];
    __shared__ float Cs[CHUNK * DSTATE];
    const int bid   = blockIdx.x;
    const int dgrp  = bid & 7;
    const int chunk = (bid >> 3) & (NCHUNK - 1);
    const int b     = bid >> 9;
    const int tid   = threadIdx.x;
    const int d     = dgrp * 256 + tid;
    const int mbase = b * SEQLEN + chunk * CHUNK;

    for (int i = tid; i < CHUNK * DSTATE; i += 256) {
        const int t = i >> 4, n = i & 15;
        Bs[i] = xdbl[(size_t)(mbase + t) * XDBL_W + DTRANK + n];
        Cs[i] = xdbl[(size_t)(mbase + t) * XDBL_W + DTRANK + DSTATE + n];
    }
    __syncthreads();

    const size_t sb = (size_t)(b * NCHUNK + chunk) * DINNER + d;
    float a[DSTATE], h[DSTATE];
#pragma unroll
    for (int n = 0; n < DSTATE; ++n) {
        a[n] = -__expf(A_log[d * DSTATE + n]);
        h[n] = H0[sb * DSTATE + n];
    }
    const float Dd = Dp[d];
    float dtv = dt[(size_t)mbase * DINNER + d];
    float xv  = xc[(size_t)mbase * DINNER + d];
    float zv  = z [(size_t)mbase * DINNER + d];
    for (int t = 0; t < CHUNK; ++t) {
        float dtn = 0.f, xn = 0.f, zn = 0.f;
        if (t + 1 < CHUNK) {                       // prefetch next timestep
            const size_t mn = (size_t)(mbase + t + 1);
            dtn = dt[mn * DINNER + d];
            xn  = xc[mn * DINNER + d];
            zn  = z [mn * DINNER + d];
        }
        const float dtx = dtv * xv;
        float yv = 0.f;
#pragma unroll
        for (int n = 0; n < DSTATE; ++n) {
            h[n] = h[n] * __expf(dtv * a[n]) + dtx * Bs[t * DSTATE + n];
            yv += h[n] * Cs[t * DSTATE + n];
        }
        yv += Dd * xv;
        yv *= zv / (1.0f + __expf(-zv));           // * silu(z)
        y[(size_t)(mbase + t) * DINNER + d] = yv;
        dtv = dtn; xv = xn; zv = zn;
    }
}

// =====================================================================
// Host-side orchestration (graph-capture safe: launches only).
// Workspace layout (floats), total ~306 MB:
//   x    [8192,2048]  @ 0           (reused as gated scan output "yin")
//   z    [8192,2048]  @ 16777216
//   xc   [8192,2048]  @ 33554432
//   xdbl [8192,  96]  @ 50331648
//   dt   [8192,2048]  @ 51118080
//   S    [2,64,2048,16] @ 67895296
//   dtsum[2,64,2048]    @ 72089600
//   H0   [2,64,2048,16] @ 72351744   (end 76546048 floats)
// =====================================================================
extern "C" void kernel_launch(void* const* d_in, const int* in_sizes, int n_in,
                              void* d_out, int out_size, void* d_ws, size_t ws_size,
                              hipStream_t stream)
{
    (void)in_sizes; (void)n_in; (void)out_size; (void)ws_size;
    const float* hs        = (const float*)d_in[0];
    const float* in_proj_w = (const float*)d_in[1];
    const float* conv_w    = (const float*)d_in[2];
    const float* conv_b    = (const float*)d_in[3];
    const float* x_proj_w  = (const float*)d_in[4];
    const float* dt_proj_w = (const float*)d_in[5];
    const float* dt_proj_b = (const float*)d_in[6];
    const float* A_log     = (const float*)d_in[7];
    const float* Dp        = (const float*)d_in[8];
    const float* out_proj_w= (const float*)d_in[9];
    float* out = (float*)d_out;
    float* ws  = (float*)d_ws;

    float* x    = ws;
    float* z    = ws + 16777216u;
    float* xc   = ws + 33554432u;
    float* xdbl = ws + 50331648u;
    float* dtb  = ws + 51118080u;
    float* S    = ws + 67895296u;
    float* dsum = ws + 72089600u;
    float* H0   = ws + 72351744u;
    float* yin  = x;                       // x is dead after the conv

    const int mblocks = MROWS / 128;       // 64

    // 1) in_proj split into x and z halves (two GEMMs, same kernel)
    {
        const int grid = mblocks * (DINNER / 64);            // 64*32 = 2048 blocks
        wmma_gemm_f32<4, 0><<<grid, 256, 0, stream>>>(
            hs, DMODEL, in_proj_w, DMODEL, x, DINNER, MROWS, DINNER, DMODEL, nullptr);
        wmma_gemm_f32<4, 0><<<grid, 256, 0, stream>>>(
            hs, DMODEL, in_proj_w + (size_t)DINNER * DMODEL, DMODEL,
            z, DINNER, MROWS, DINNER, DMODEL, nullptr);
    }
    // 2) causal conv + SiLU
    conv_silu_kernel<<<(MROWS * DINNER) / 256, 256, 0, stream>>>(x, conv_w, conv_b, xc);
    // 3) x_proj -> x_dbl [8192, 96]  (NT=2 -> 3 n-blocks)
    {
        const int grid = mblocks * (XDBL_W / 32);            // 64*3 = 192
        wmma_gemm_f32<2, 0><<<grid, 256, 0, stream>>>(
            xc, DINNER, x_proj_w, DINNER, xdbl, XDBL_W, MROWS, XDBL_W, DINNER, nullptr);
    }
    // 4) dt = softplus(dt_lo @ dt_proj_w^T + b)   (A = x_dbl[:, :64], lda=96)
    {
        const int grid = mblocks * (DINNER / 64);            // 2048
        wmma_gemm_f32<4, 1><<<grid, 256, 0, stream>>>(
            xdbl, XDBL_W, dt_proj_w, DTRANK, dtb, DINNER, MROWS, DINNER, DTRANK, dt_proj_b);
    }
    // 5) chunked selective scan (3 phases), fused gating in phase 3
    scan_phase1<<<BATCH * NCHUNK * (DINNER / 256), 256, 0, stream>>>(xc, dtb, xdbl, A_log, S, dsum);
    scan_phase2<<<(BATCH * DINNER * DSTATE) / 256, 256, 0, stream>>>(S, dsum, A_log, H0);
    scan_phase3<<<BATCH * NCHUNK * (DINNER / 256), 256, 0, stream>>>(xc, dtb, xdbl, A_log, Dp, z, H0, yin);
    // 6) out_proj -> d_out [8192, 1024]
    {
        const int grid = mblocks * (DMODEL / 64);            // 1024
        wmma_gemm_f32<4, 0><<<grid, 256, 0, stream>>>(
            yin, DINNER, out_proj_w, DINNER, out, DMODEL, MROWS, DMODEL, DINNER, nullptr);
    }
}